// HeatLoss_NextGen_3_44032004718833
// MI455X (gfx1250) — compile-verified
//
#include <hip/hip_runtime.h>

// ---------------------------------------------------------------------------
// Masked L1 loss over [B=16,F=16,H=256,W=256] fp32.
// Memory-bound: 144 MB traffic -> ~6.2 us floor at 23.3 TB/s.
//
// Lane l owns feature f=l&15. Lanes 0-15 process the first half of the
// wave's 128-position chunk (K-columns {0,1} of the WMMA A-matrix),
// lanes 16-31 the second half (K-columns {2,3}) -- matching the documented
// 16x4 f32 A layout while keeping each lane's memory stream contiguous
// (float4 loads, one b128 mask load per 8 pair-steps, immediate offsets).
// Five v8f WMMA accumulators (sd, s1, s3, c1, c3) do all per-feature
// reductions via D += A x ones(4x16); D rows are exact fp32 row sums read
// from lane 0 / lane 16. m_all comes from one wave32 ballot per position,
// turned per-lane via a precomputed halfmask (3 VALU).
// 1024 blocks x 256 threads = 8192 waves for load-latency hiding.
// Deterministic two-pass reduction (block partials in d_ws, no fp atomics).
// ---------------------------------------------------------------------------

typedef __attribute__((ext_vector_type(2))) float v2f;
typedef __attribute__((ext_vector_type(8))) float v8f;

namespace {
constexpr int kNF     = 16;          // features
constexpr int kPlane  = 256 * 256;   // H*W
constexpr int kNB     = 16;          // batch
constexpr int kNPos   = kNB * kPlane;                    // 1,048,576 positions
constexpr int kBlocks = 1024;
constexpr int kTPB    = 256;
constexpr int kWavesPerBlock = kTPB / 32;                // 8 (wave32)
constexpr int kTotalWaves    = kBlocks * kWavesPerBlock; // 8192
constexpr int kChunk  = kNPos / kTotalWaves;             // 128 positions / wave
constexpr int kHalf   = kChunk / 2;                      // 64 per lane-half
constexpr int kOuter  = kHalf / 16;                      // 4: 16 floats/lane/outer
constexpr int kSlots  = 65; // [0,16) sd | [16,32) s1 | [32,48) s3 | [48,64) c1 | 64 c3
}

__global__ __launch_bounds__(kTPB)
void heatloss_partial(const float* __restrict__ inp,
                      const float* __restrict__ tgt,
                      const unsigned char* __restrict__ msk,
                      float* __restrict__ gpart)
{
  const int lane  = threadIdx.x & 31;
  const int wib   = threadIdx.x >> 5;
  const int wave  = blockIdx.x * kWavesPerBlock + wib;
  const int f     = lane & 15;
  const bool upper = lane >= 16;
  const unsigned halfmask = upper ? 0xFFFF0000u : 0x0000FFFFu;

  // Chunk base: 128-aligned -> never crosses a 65536-position plane.
  const int cb   = wave * kChunk;
  const int b    = cb >> 16;
  const int off  = cb & (kPlane - 1);
  const int base = (b * kNF + f) * kPlane + off + (upper ? kHalf : 0);

  const float*         pi = inp + base;
  const float*         pt = tgt + base;
  const unsigned char* pm = msk + base;

  v8f acc_sd = {}, acc_s1 = {}, acc_s3 = {}, acc_c1 = {}, acc_c3 = {};
  v2f bones; bones.x = 1.0f; bones.y = 1.0f;   // all-ones B: layout-independent

  for (int oi = 0; oi < kOuter; ++oi) {
    v2f va_sd = {}, va_s1 = {}, va_s3 = {}, va_c1 = {}, va_c3 = {};

    __builtin_prefetch(pi + oi * 16 + 16, 0, 0);   // global_prefetch_b8
    __builtin_prefetch(pt + oi * 16 + 16, 0, 0);

    const uint4 mq = *(const uint4*)(pm + oi * 16);  // 16 mask bytes = 8 pairs
#pragma unroll
    for (int j = 0; j < 4; ++j) {
      const float4 x = *(const float4*)(pi + oi * 16 + j * 4);
      const float4 y = *(const float4*)(pt + oi * 16 + j * 4);
      const unsigned mw = (j == 0) ? mq.x : (j == 1) ? mq.y : (j == 2) ? mq.z : mq.w;

#pragma unroll
      for (int h = 0; h < 2; ++h) {              // two position-pairs per float4
        const float xa = h ? x.z : x.x, xb = h ? x.w : x.y;
        const float ya = h ? y.z : y.x, yb = h ? y.w : y.y;
        const unsigned byt0 = (mw >> (h * 16)) & 0xFFu;
        const unsigned byt1 = (mw >> (h * 16 + 8)) & 0xFFu;

        const float d0 = __builtin_fabsf(xa - ya);
        const float d1 = __builtin_fabsf(xb - yb);
        const bool mb0 = byt0 != 0u;
        const bool mb1 = byt1 != 0u;
        const float m0 = mb0 ? 1.0f : 0.0f;
        const float m1 = mb1 ? 1.0f : 0.0f;

        // bits 0-15: 16 feature masks at the lo-half position;
        // bits 16-31: at the hi-half position. Select own half via halfmask.
        const unsigned bal0 = (unsigned)__ballot(mb0);
        const unsigned bal1 = (unsigned)__ballot(mb1);
        const float a0 = (bal0 & halfmask) ? 1.0f : 0.0f;
        const float a1 = (bal1 & halfmask) ? 1.0f : 0.0f;

        va_sd.x += d0;                      va_sd.y += d1;
        va_s1.x  = __builtin_fmaf(d0, m0, va_s1.x);
        va_s1.y  = __builtin_fmaf(d1, m1, va_s1.y);
        va_s3.x  = __builtin_fmaf(d0, a0, va_s3.x);
        va_s3.y  = __builtin_fmaf(d1, a1, va_s3.y);
        va_c1.x += m0;                      va_c1.y += m1;
        va_c3.x += a0;                      va_c3.y += a1;
      }
    }
    // D[f][*] += va(lane f).x + va(lane f).y + va(lane f+16).x + va(lane f+16).y
    acc_sd = __builtin_amdgcn_wmma_f32_16x16x4_f32(false, va_sd, false, bones, (short)0, acc_sd, false, false);
    acc_s1 = __builtin_amdgcn_wmma_f32_16x16x4_f32(false, va_s1, false, bones, (short)0, acc_s1, false, false);
    acc_s3 = __builtin_amdgcn_wmma_f32_16x16x4_f32(false, va_s3, false, bones, (short)0, acc_s3, false, false);
    acc_c1 = __builtin_amdgcn_wmma_f32_16x16x4_f32(false, va_c1, false, bones, (short)0, acc_c1, false, false);
    acc_c3 = __builtin_amdgcn_wmma_f32_16x16x4_f32(false, va_c3, false, bones, (short)0, acc_c3, false, false);
  }

  // C/D layout: VGPR r, lanes 0-15 -> M=r (f=0..7, col=lane); lanes 16-31 -> M=8+r.
  // Row sums are replicated across columns; every row of acc_c3 equals the
  // wave's total m_all count.
  __shared__ float ls[kWavesPerBlock][kSlots + 7];
  if (lane == 0 || lane == 16) {
    const int fb = upper ? 8 : 0;
#pragma unroll
    for (int r = 0; r < 8; ++r) {
      ls[wib][ 0 + fb + r] = acc_sd[r];
      ls[wib][16 + fb + r] = acc_s1[r];
      ls[wib][32 + fb + r] = acc_s3[r];
      ls[wib][48 + fb + r] = acc_c1[r];
    }
    if (lane == 0) ls[wib][64] = acc_c3[0];
  }
  __syncthreads();
  if (threadIdx.x < kSlots) {
    float s = 0.0f;
    for (int w = 0; w < kWavesPerBlock; ++w) s += ls[w][threadIdx.x];
    gpart[blockIdx.x * kSlots + threadIdx.x] = s;   // fully overwritten each launch
  }
}

__global__ void heatloss_finalize(const float* __restrict__ gpart,
                                  float* __restrict__ out)
{
  __shared__ float tot[kSlots];
  const int t = threadIdx.x;
  if (t < kSlots) {
    float s = 0.0f;
    for (int b = 0; b < kBlocks; ++b) s += gpart[b * kSlots + t];
    tot[t] = s;
  }
  __syncthreads();
  if (t == 0) {
    const float Ntot = (float)kNPos;
    const float c3 = tot[64];
    float l1 = 0.0f, l2 = 0.0f, l3 = 0.0f;
    for (int ff = 0; ff < kNF; ++ff) {
      const float sd = tot[ff], s1 = tot[16 + ff], s3 = tot[32 + ff], c1 = tot[48 + ff];
      const float s2 = sd - s1;
      const float c2 = Ntot - c1;
      l1 += (c1 > 0.0f) ? s1 / fmaxf(c1, 1.0f) : 0.0f;   // loss_features
      l2 += (c2 > 0.0f) ? s2 / fmaxf(c2, 1.0f) : 0.0f;   // loss_background
      l3 += (c3 > 0.0f) ? s3 / fmaxf(c3, 1.0f) : 0.0f;   // loss_all_features
    }
    out[0] = (l1 / 16.0f + l3 / 16.0f + l2 / 16.0f) / 3.0f;
  }
}

extern "C" void kernel_launch(void* const* d_in, const int* in_sizes, int n_in,
                              void* d_out, int out_size, void* d_ws, size_t ws_size,
                              hipStream_t stream) {
  const float*         inp = (const float*)d_in[0];
  const float*         tgt = (const float*)d_in[1];
  const unsigned char* msk = (const unsigned char*)d_in[2];  // jnp.bool_: 1 byte/elem
  float* gpart = (float*)d_ws;   // needs kBlocks*kSlots*4 = 266,240 bytes

  heatloss_partial<<<kBlocks, kTPB, 0, stream>>>(inp, tgt, msk, gpart);
  heatloss_finalize<<<1, 128, 0, stream>>>(gpart, (float*)d_out);
}